// FactorizedSpectralConv_15650860827414
// MI455X (gfx1250) — compile-verified
//
#include <hip/hip_runtime.h>
#include <math.h>

// ---------------------------------------------------------------------------
// FactorizedSpectralConv (spherical harmonic spectral conv) for MI455X gfx1250
//
// All heavy stages are affine-strided batched GEMMs on V_WMMA_F32_16X16X4_F32
// (fp32 in/out, matching reference fp32 semantics). ~107 GFLOP total vs
// ~0.3 GB touched -> purely matrix-unit bound, so everything rides WMMA.
// Global->LDS loads are software-pipelined one K-tile ahead through registers;
// both LDS operands are stored K-contiguous so fragments load as ds_load_b64.
// ---------------------------------------------------------------------------

typedef __attribute__((ext_vector_type(2))) float v2f;
typedef __attribute__((ext_vector_type(8))) float v8f;

#define NLAT 181
#define NLON 360
#define LMAX 181
#define MMAX 181
#define NB   2
#define CIN  256
#define COUT 256

#define PI_D 3.14159265358979323846

// -------------------- generic batched strided WMMA GEMM --------------------
// C[z][r,c] = alpha * sum_k A[z][r,k]*B[z][k,c]  (+ C_old if accum)
// Operands addressed as base + z*BS + row*RS + col*CS (element strides).
// Block tile 64x128, K-tile 16; 8 waves (4 row-slabs x 2 col-slabs); each
// wave computes 16x64 via four v8f accumulators sharing one A fragment.
#define BM 64
#define BN 128
#define KT 16
#define KP (KT + 2)   // even pad: 8B-aligned v2f frags, conflict-free banks

__global__ __launch_bounds__(256) void gemm_wmma_f32(
    const float* __restrict__ A, const float* __restrict__ B,
    float* __restrict__ C,
    int M, int N, int K,
    long aRS, long aCS, long aBS,
    long bRS, long bCS, long bBS,
    long cRS, long cCS, long cBS,
    float alpha, int accum)
{
    __shared__ float As[BM][KP];   // [row][k]  (k contiguous)
    __shared__ float Bs[BN][KP];   // [col][k]  (k contiguous)

    const int tid  = threadIdx.x;
    const int lane = tid & 31;
    const int wid  = tid >> 5;          // 8 waves
    const int wR   = wid >> 1;          // 0..3 -> 16-row slab
    const int wC   = wid & 1;           // 0..1 -> 64-col slab
    const int half = lane >> 4;         // 0/1  (K-split half per ISA layout)
    const int lr   = lane & 15;

    const long z = blockIdx.z;
    const float* __restrict__ Ab = A + z * aBS;
    const float* __restrict__ Bb = B + z * bBS;
    float* __restrict__       Cb = C + z * cBS;

    const int m0 = blockIdx.y * BM;
    const int n0 = blockIdx.x * BN;

    // ---- per-thread load descriptors (clamped => always-safe addresses) ---
    // A tile: BM*KT = 1024 elems, 4 per thread.  B tile: KT*BN = 2048, 8 each.
    long aOff[4]; bool aOk[4]; int akk[4]; long arC[4];
    long bOff[8]; bool bOk[8]; int bkk[8]; long bcC[8];
#pragma unroll
    for (int e = 0; e < 4; ++e) {
        int idx = tid + e * 256;
        int r = idx >> 4, kk = idx & 15;        // A: row, k
        int gr = m0 + r;
        aOk[e] = gr < M;
        arC[e] = aOk[e] ? gr : (M - 1);
        akk[e] = kk;
        aOff[e] = arC[e] * aRS + (long)kk * aCS;
    }
#pragma unroll
    for (int e = 0; e < 8; ++e) {
        int idx = tid + e * 256;
        int bk = idx >> 7, j = idx & 127;       // B: k, col
        int gc = n0 + j;
        bOk[e] = gc < N;
        bcC[e] = bOk[e] ? gc : (N - 1);
        bkk[e] = bk;
        bOff[e] = (long)bk * bRS + bcC[e] * bCS;
    }
    const long aStep = (long)KT * aCS;
    const long bStep = (long)KT * bRS;

    float va[4], vb[8];
    // load one K-tile into registers (issued early; zero-filled OOB)
    auto load_regs = [&](int k0) {
        if (k0 + KT <= K) {                       // steady state: no k guards
#pragma unroll
            for (int e = 0; e < 4; ++e) va[e] = Ab[aOff[e]];
#pragma unroll
            for (int e = 0; e < 8; ++e) vb[e] = Bb[bOff[e]];
#pragma unroll
            for (int e = 0; e < 4; ++e) if (!aOk[e]) va[e] = 0.f;
#pragma unroll
            for (int e = 0; e < 8; ++e) if (!bOk[e]) vb[e] = 0.f;
        } else {                                  // K tail: clamped readdress
#pragma unroll
            for (int e = 0; e < 4; ++e) {
                int gk = k0 + akk[e];
                int gkc = gk < K ? gk : (K - 1);
                va[e] = Ab[arC[e] * aRS + (long)gkc * aCS];
                if (!(aOk[e] && gk < K)) va[e] = 0.f;
            }
#pragma unroll
            for (int e = 0; e < 8; ++e) {
                int gk = k0 + bkk[e];
                int gkc = gk < K ? gk : (K - 1);
                vb[e] = Bb[(long)gkc * bRS + bcC[e] * bCS];
                if (!(bOk[e] && gk < K)) vb[e] = 0.f;
            }
        }
#pragma unroll
        for (int e = 0; e < 4; ++e) aOff[e] += aStep;
#pragma unroll
        for (int e = 0; e < 8; ++e) bOff[e] += bStep;
    };
    auto stash_lds = [&]() {
#pragma unroll
        for (int e = 0; e < 4; ++e) {
            int idx = tid + e * 256;
            As[idx >> 4][idx & 15] = va[e];      // [row][k]
        }
#pragma unroll
        for (int e = 0; e < 8; ++e) {
            int idx = tid + e * 256;
            Bs[idx & 127][idx >> 7] = vb[e];     // [col][k]
        }
    };

    v8f acc[4];
#pragma unroll
    for (int t = 0; t < 4; ++t)
        acc[t] = (v8f){0.f, 0.f, 0.f, 0.f, 0.f, 0.f, 0.f, 0.f};

    auto compute_tile = [&]() {
#pragma unroll
        for (int kk = 0; kk < KT; kk += 4) {
            // A 16x4: lanes0-15 -> K=0,1 ; lanes16-31 -> K=2,3 (contig pair)
            v2f a = *(const v2f*)&As[wR * 16 + lr][kk + half * 2];
#pragma unroll
            for (int t = 0; t < 4; ++t) {
                // B 4x16 mirrors A; four 16-col tiles share the A fragment
                v2f b = *(const v2f*)&Bs[wC * 64 + t * 16 + lr][kk + half * 2];
                acc[t] = __builtin_amdgcn_wmma_f32_16x16x4_f32(
                    false, a, false, b, (short)0, acc[t], false, false);
            }
        }
    };

    // -------- software-pipelined main loop: load k+1 while computing k -----
    load_regs(0);
    for (int k0 = 0; k0 < K; k0 += KT) {
        stash_lds();
        __syncthreads();
        if (k0 + KT < K) load_regs(k0 + KT);   // overlap with WMMA below
        compute_tile();
        __syncthreads();
    }

    // C/D 16x16 layout: VGPR j -> row j (lanes0-15) / row j+8 (lanes16-31)
#pragma unroll
    for (int t = 0; t < 4; ++t) {
#pragma unroll
        for (int j = 0; j < 8; ++j) {
            int gr = m0 + wR * 16 + half * 8 + j;
            int gc = n0 + wC * 64 + t * 16 + lr;
            if (gr < M && gc < N) {
                long off = (long)gr * cRS + (long)gc * cCS;
                float v = alpha * acc[t][j];
                if (accum) v += Cb[off];
                Cb[off] = v;
            }
        }
    }
}

// ------------------ Legendre tables (fp64 recurrence, fp32 out) ------------
// pct [l][m][k], pctq = pct * quad[k]   (quad folded into forward transform)
__global__ void precompute_pct(float* __restrict__ pct, float* __restrict__ pctq)
{
    int t = blockIdx.x * blockDim.x + threadIdx.x;
    if (t >= MMAX * NLAT) return;
    int m = t / NLAT;
    int k = t - m * NLAT;

    double theta = PI_D * (k + 0.5) / NLAT;
    double ct = cos(theta), st = sin(theta);
    float quad = (float)(st * (PI_D / NLAT) * (2.0 * PI_D / NLON));

    double pmm = 1.0 / sqrt(4.0 * PI_D);
    for (int mm = 1; mm <= m; ++mm)
        pmm *= sqrt((2.0 * mm + 1.0) / (2.0 * mm)) * st;

    const long MS = (long)MMAX * NLAT;
    for (int l = 0; l < m; ++l) {
        long o = (long)l * MS + (long)m * NLAT + k;
        pct[o] = 0.f; pctq[o] = 0.f;
    }
    double pl2 = pmm;  // l = m
    {
        long o = (long)m * MS + (long)m * NLAT + k;
        pct[o] = (float)pl2; pctq[o] = (float)pl2 * quad;
    }
    if (m + 1 < LMAX) {
        double pl1 = sqrt(2.0 * m + 3.0) * ct * pmm;
        long o = (long)(m + 1) * MS + (long)m * NLAT + k;
        pct[o] = (float)pl1; pctq[o] = (float)pl1 * quad;
        for (int l = m + 2; l < LMAX; ++l) {
            double ld = (double)l, md = (double)m, l1 = ld - 1.0;
            double a = sqrt((4.0 * ld * ld - 1.0) / (ld * ld - md * md));
            double b = sqrt((l1 * l1 - md * md) / (4.0 * l1 * l1 - 1.0));
            double pl = a * (ct * pl1 - b * pl2);
            long o2 = (long)l * MS + (long)m * NLAT + k;
            pct[o2] = (float)pl; pctq[o2] = (float)pl * quad;
            pl2 = pl1; pl1 = pl;
        }
    }
}

// ------------------ DFT bases (rfft fwd, irfft*nlon inverse) ---------------
// fwd:  Xr = sum_n x*cos(2pi m n/N), Xi = sum_n x*(-sin)
// inv:  out[n] = sum_m c_m*(Xr*cos - Xi*sin), c_m = 1 for m=0,180 else 2
__global__ void precompute_basis(float* __restrict__ Ac, float* __restrict__ Asn,
                                 float* __restrict__ Er, float* __restrict__ Ei)
{
    int t = blockIdx.x * blockDim.x + threadIdx.x;
    if (t >= MMAX * NLON) return;
    int m = t / NLON;
    int n = t - m * NLON;
    int r = (m * n) % NLON;
    double ang = 2.0 * PI_D * (double)r / (double)NLON;
    double c = cos(ang), s = sin(ang);
    Ac[t]  = (float)c;
    Asn[t] = (float)(-s);
    double cm = (m == 0 || m == NLON / 2) ? 1.0 : 2.0;
    Er[t] = (float)(cm * c);
    Ei[t] = (float)(-cm * s);
}

// ---------------------------------------------------------------------------
extern "C" void kernel_launch(void* const* d_in, const int* in_sizes, int n_in,
                              void* d_out, int out_size, void* d_ws, size_t ws_size,
                              hipStream_t stream)
{
    const float* x  = (const float*)d_in[0];   // [2,256,181,360]
    const float* wr = (const float*)d_in[1];   // [256,256,181]
    const float* wi = (const float*)d_in[2];   // [256,256,181]
    float* out = (float*)d_out;
    float* ws  = (float*)d_ws;

    const long PCT = (long)LMAX * MMAX * NLAT;        // 5,929,741
    const long BAS = (long)MMAX * NLON;               // 65,160
    const long BIG = (long)MMAX * NLAT * (NB * CIN);  // 16,773,632
    const long OUTE = (long)NB * COUT * NLAT * NLON;  // 33,361,920

    float* pct  = ws;
    float* pctq = pct + PCT;
    float* Ac   = pctq + PCT;
    float* Asn  = Ac + BAS;
    float* Er   = Asn + BAS;
    float* Ei   = Er + BAS;
    float* bigA = Ei + BAS;       // XF_re  -> reused as XP_re
    float* bigB = bigA + BIG;     // XF_im  -> reused as XP_im
    float* bigC = bigB + BIG;     // CO_re  -> reused as XS_re
    float* bigD = bigC + BIG;     // CO_im  -> reused as XS_im

    dim3 blk(256);
    precompute_pct<<<(MMAX * NLAT + 255) / 256, blk, 0, stream>>>(pct, pctq);
    precompute_basis<<<(MMAX * NLON + 255) / 256, blk, 0, stream>>>(Ac, Asn, Er, Ei);

    const long BC = NB * CIN;           // 512
    const long MBC = (long)MMAX * BC;   // 92,672  (l / bo stride of big buffers)
    const long MK  = (long)MMAX * NLAT; // 32,761

    // -------- Stage A: DFT over longitude, batched over latitude k ---------
    // XF[m][k][bc] = sum_n basis[m][n] * x[bc][k][n]
    {
        dim3 g((BC + BN - 1) / BN, (MMAX + BM - 1) / BM, NLAT);
        gemm_wmma_f32<<<g, blk, 0, stream>>>(Ac,  x, bigA, MMAX, (int)BC, NLON,
            NLON, 1, 0,   1, (long)NLAT * NLON, NLON,   MBC, 1, BC,  1.f, 0);
        gemm_wmma_f32<<<g, blk, 0, stream>>>(Asn, x, bigB, MMAX, (int)BC, NLON,
            NLON, 1, 0,   1, (long)NLAT * NLON, NLON,   MBC, 1, BC,  1.f, 0);
    }

    // -------- Stage B: forward Legendre (quad folded), batched over m ------
    // CO[l][m][bc] = sum_k pctq[l][m][k] * XF[m][k][bc]
    {
        dim3 g((BC + BN - 1) / BN, (LMAX + BM - 1) / BM, MMAX);
        gemm_wmma_f32<<<g, blk, 0, stream>>>(pctq, bigA, bigC, LMAX, (int)BC, NLAT,
            MK, 1, NLAT,   BC, 1, MBC,   MBC, 1, BC,  1.f, 0);
        gemm_wmma_f32<<<g, blk, 0, stream>>>(pctq, bigB, bigD, LMAX, (int)BC, NLAT,
            MK, 1, NLAT,   BC, 1, MBC,   MBC, 1, BC,  1.f, 0);
    }

    // -------- Stage C: dhconv, complex per-l channel mixing, batch over l --
    // XP[l][m][b][o] = sum_i w[i][o][l] * CO[l][m][b][i]
    // column index j = 2m+b has single stride 256 in the [l][m][b][c] layout.
    {
        const int Nj = NB * MMAX; // 362
        dim3 g((Nj + BN - 1) / BN, (COUT + BM - 1) / BM, LMAX);
        const long aRS = LMAX, aCS = (long)COUT * LMAX, aBS = 1;
        // xp_re = wr*co_re - wi*co_im
        gemm_wmma_f32<<<g, blk, 0, stream>>>(wr, bigC, bigA, COUT, Nj, CIN,
            aRS, aCS, aBS,   1, CIN, MBC,   1, CIN, MBC,   1.f, 0);
        gemm_wmma_f32<<<g, blk, 0, stream>>>(wi, bigD, bigA, COUT, Nj, CIN,
            aRS, aCS, aBS,   1, CIN, MBC,   1, CIN, MBC,  -1.f, 1);
        // xp_im = wi*co_re + wr*co_im
        gemm_wmma_f32<<<g, blk, 0, stream>>>(wi, bigC, bigB, COUT, Nj, CIN,
            aRS, aCS, aBS,   1, CIN, MBC,   1, CIN, MBC,   1.f, 0);
        gemm_wmma_f32<<<g, blk, 0, stream>>>(wr, bigD, bigB, COUT, Nj, CIN,
            aRS, aCS, aBS,   1, CIN, MBC,   1, CIN, MBC,   1.f, 1);
    }

    // -------- Stage D: Legendre synthesis, batched over m ------------------
    // XS[bo][k][m] = sum_l pct[l][m][k] * XP[l][m][bo]
    {
        const long BO = NB * COUT; // 512
        dim3 g((BO + BN - 1) / BN, (NLAT + BM - 1) / BM, MMAX);
        gemm_wmma_f32<<<g, blk, 0, stream>>>(pct, bigA, bigC, NLAT, (int)BO, LMAX,
            1, MK, NLAT,   MBC, 1, BO,   MMAX, MK, 1,   1.f, 0);
        gemm_wmma_f32<<<g, blk, 0, stream>>>(pct, bigB, bigD, NLAT, (int)BO, LMAX,
            1, MK, NLAT,   MBC, 1, BO,   MMAX, MK, 1,   1.f, 0);
    }

    // -------- Stage E: inverse DFT (irfft * nlon folded), single GEMM ------
    // out[(b,o,k)][n] = sum_m XSr*Er[m][n] + XSi*Ei[m][n]
    {
        const int Mrows = (int)(NB * COUT * NLAT); // 92,672
        dim3 g((NLON + BN - 1) / BN, (Mrows + BM - 1) / BM, 1);
        gemm_wmma_f32<<<g, blk, 0, stream>>>(bigC, Er, out, Mrows, NLON, MMAX,
            MMAX, 1, 0,   NLON, 1, 0,   NLON, 1, 0,   1.f, 0);
        gemm_wmma_f32<<<g, blk, 0, stream>>>(bigD, Ei, out, Mrows, NLON, MMAX,
            MMAX, 1, 0,   NLON, 1, 0,   NLON, 1, 0,   1.f, 1);
    }

    // -------- residual = x (second output of the tuple) --------------------
    (void)hipMemcpyAsync(out + OUTE, x, OUTE * sizeof(float),
                         hipMemcpyDeviceToDevice, stream);
}